// MultiHeadQGFDLayer_31104153157929
// MI455X (gfx1250) — compile-verified
//
#include <hip/hip_runtime.h>
#include <hip/hip_bf16.h>

typedef float v2f __attribute__((ext_vector_type(2)));
typedef float v8f __attribute__((ext_vector_type(8)));

// ---------------------------------------------------------------------------
// Generic batched fp32 WMMA GEMM:
//   C[b] = scale * (A[b] @ B[b](^T)) + add_scale * Add[b] + bias
// Block = 256 threads (8 waves), block tile 256x64 (wave grid 4x2, 64x32/wave,
// 8 v_wmma_f32_16x16x4_f32 per k-step per wave).
// Requires M % 256 == 0, N % 64 == 0, K % 4 == 0 (true for every call here),
// so there is no divergence and EXEC is all-ones at every WMMA (ISA 7.12 req).
// Per-batch sub-matrices are < 2^31 elements, so all hot-loop addressing is
// uniform-base + 32-bit lane offset (GVS saddr mode, minimal VALU).
// ---------------------------------------------------------------------------
template<bool TRANSB, bool HAS_ADD, bool HAS_BIAS>
__global__ __launch_bounds__(256)
void wmma_gemm_f32(const float* __restrict__ A,
                   const float* __restrict__ B,
                   const float* Add,               // may alias C (elementwise)
                   const float* __restrict__ bias,
                   float* C,
                   int K, int lda, int ldb, int ldc,
                   long long saB, long long sbB, long long scB, long long sAddB,
                   float scale, float add_scale) {
  const int lane = threadIdx.x & 31;
  const int wave = threadIdx.x >> 5;
  const long long bz = blockIdx.z;

  A += bz * saB;
  B += bz * sbB;
  C += bz * scB;
  if (HAS_ADD) Add += bz * sAddB;

  const int wm = blockIdx.x * 256 + (wave & 3) * 64;  // wave row origin
  const int wn = blockIdx.y * 64 + (wave >> 2) * 32;  // wave col origin

  const v8f zero = {0.f, 0.f, 0.f, 0.f, 0.f, 0.f, 0.f, 0.f};
  v8f acc[4][2] = {{zero, zero}, {zero, zero}, {zero, zero}, {zero, zero}};

  // A 16x4 fragment mapping (ISA 7.12.2): lanes 0-15 -> K={0,1}, 16-31 -> K={2,3}
  const int ar = lane & 15;          // row within 16-row tile
  const int ak = (lane >> 4) << 1;   // k offset (0 or 2)
  const int bc = lane & 15;          // col within 16-col tile

  unsigned aoff = (unsigned)(wm + ar) * (unsigned)lda + (unsigned)ak;
  unsigned boff = TRANSB ? (unsigned)(wn + bc) * (unsigned)ldb + (unsigned)ak
                         : (unsigned)ak * (unsigned)ldb + (unsigned)(wn + bc);
  const unsigned bstep = TRANSB ? 4u : 4u * (unsigned)ldb;
  const unsigned arow16 = 16u * (unsigned)lda;
  const unsigned brow16 = 16u * (unsigned)ldb;

#pragma unroll 2
  for (int k = 0; k < K; k += 4) {
    v2f a[4], b[2];
#pragma unroll
    for (int i = 0; i < 4; i++)
      a[i] = *(const v2f*)(A + aoff + (unsigned)i * arow16);
    if (TRANSB) {
      b[0] = *(const v2f*)(B + boff);
      b[1] = *(const v2f*)(B + boff + brow16);
    } else {
      b[0].x = B[boff];
      b[0].y = B[boff + (unsigned)ldb];
      b[1].x = B[boff + 16u];
      b[1].y = B[boff + (unsigned)ldb + 16u];
    }
    aoff += 4u;
    boff += bstep;
#pragma unroll
    for (int i = 0; i < 4; i++) {
      acc[i][0] = __builtin_amdgcn_wmma_f32_16x16x4_f32(false, a[i], false, b[0], (short)0, acc[i][0], false, false);
      acc[i][1] = __builtin_amdgcn_wmma_f32_16x16x4_f32(false, a[i], false, b[1], (short)0, acc[i][1], false, false);
    }
  }

  // C/D 16x16 layout: VGPR r, lanes 0-15: M=r; lanes 16-31: M=8+r
  const int cn = lane & 15;
  const int mhalf = (lane >> 4) * 8;
#pragma unroll
  for (int ti = 0; ti < 4; ti++) {
#pragma unroll
    for (int tj = 0; tj < 2; tj++) {
      const int row0 = wm + ti * 16 + mhalf;
      const int col = wn + tj * 16 + cn;
      const float bval = HAS_BIAS ? bias[col] : 0.f;
#pragma unroll
      for (int r = 0; r < 8; r++) {
        const long long idx = (long long)(row0 + r) * ldc + col;
        float v = scale * acc[ti][tj][r] + bval;
        if (HAS_ADD) v += add_scale * Add[idx];
        C[idx] = v;
      }
    }
  }
}

// ---------------------------------------------------------------------------
// In-place row softmax: one 256-thread block per row.
// ---------------------------------------------------------------------------
__global__ __launch_bounds__(256)
void row_softmax(float* __restrict__ buf, int cols) {
  __shared__ float red[8];
  const int tid = threadIdx.x;
  const int lane = tid & 31;
  const int wave = tid >> 5;
  float* row = buf + (long long)blockIdx.x * cols;

  float m = -3.402823466e38f;
  for (int i = tid; i < cols; i += 256) m = fmaxf(m, row[i]);
#pragma unroll
  for (int off = 16; off; off >>= 1) m = fmaxf(m, __shfl_xor(m, off, 32));
  if (lane == 0) red[wave] = m;
  __syncthreads();
  float mAll = red[0];
#pragma unroll
  for (int w = 1; w < 8; w++) mAll = fmaxf(mAll, red[w]);
  __syncthreads();

  float s = 0.f;
  for (int i = tid; i < cols; i += 256) {
    float e = __expf(row[i] - mAll);
    row[i] = e;
    s += e;
  }
#pragma unroll
  for (int off = 16; off; off >>= 1) s += __shfl_xor(s, off, 32);
  if (lane == 0) red[wave] = s;
  __syncthreads();
  float tot = 0.f;
#pragma unroll
  for (int w = 0; w < 8; w++) tot += red[w];
  const float inv = 1.0f / tot;
  for (int i = tid; i < cols; i += 256) row[i] *= inv;
}

// ---------------------------------------------------------------------------
// Async-engine copy (p2 -> d_out p-region): global -> LDS -> global using the
// CDNA5 ASYNCcnt-tracked path (ISA 10 / 15.18.3). Each thread owns a private
// 16-byte LDS slot, so no barriers are required, only asynccnt waits:
//   load(async, LDS) ; wait asynccnt 0 ; store(async, from LDS) ; wait.
// ---------------------------------------------------------------------------
__global__ __launch_bounds__(256)
void async_copy_f32x4(const float* __restrict__ src, float* __restrict__ dst,
                      long long n4) {
  __shared__ float tile[256 * 4];
  const int tid = threadIdx.x;
  // Generic address of an LDS object: low 32 bits are the LDS byte offset.
  const unsigned ldsOff = (unsigned)(size_t)(&tile[tid * 4]);

  long long i = (long long)blockIdx.x * blockDim.x + tid;
  const long long stride = (long long)gridDim.x * blockDim.x;
  for (; i < n4; i += stride) {
    const float* gsrc = src + i * 4;
    float* gdst = dst + i * 4;
    asm volatile("global_load_async_to_lds_b128 %0, %1, off"
                 :: "v"(ldsOff), "v"(gsrc) : "memory");
    asm volatile("s_wait_asynccnt 0x0" ::: "memory");
    asm volatile("global_store_async_from_lds_b128 %0, %1, off"
                 :: "v"(gdst), "v"(ldsOff) : "memory");
    asm volatile("s_wait_asynccnt 0x0" ::: "memory");
  }
}

extern "C" void kernel_launch(void* const* d_in, const int* in_sizes, int n_in,
                              void* d_out, int out_size, void* d_ws, size_t ws_size,
                              hipStream_t stream) {
  (void)in_sizes; (void)n_in; (void)out_size; (void)ws_size;

  const float* X  = (const float*)d_in[0];
  const float* Wq = (const float*)d_in[1];
  const float* bq = (const float*)d_in[2];
  const float* Wk = (const float*)d_in[3];
  const float* bk = (const float*)d_in[4];
  const float* Wv = (const float*)d_in[5];
  const float* bv = (const float*)d_in[6];
  const float* Wo = (const float*)d_in[7];
  const float* bo = (const float*)d_in[8];

  const int N = 2048, D = 1024, H = 16, HD = 64;
  const long long NN = (long long)N * N;  // 4,194,304 per head

  float* out_main = (float*)d_out;                 // [N, D]
  float* out_p    = out_main + (long long)N * D;   // [H, N, N]

  float* ws = (float*)d_ws;
  float* Q  = ws;                                  // [N, D]
  float* Kb = Q + (long long)N * D;                // [N, D]
  float* V  = Kb + (long long)N * D;               // [N, D]
  float* At = V + (long long)N * D;                // [N, D] attn pre-proj
  float* p0 = At + (long long)N * D;               // [H, N, N]
  float* Pm = p0 + (long long)H * NN;              // [H, N, N]

  const dim3 blk(256);
  const dim3 gproj(N / 256, D / 64, 1);   // (8,16,1)
  const dim3 gsc(N / 256, N / 64, H);     // (8,32,16)
  const dim3 gat(N / 256, HD / 64, H);    // (8,1,16)

  // 1) Projections: Q/K/V = X @ W + b
  wmma_gemm_f32<false, false, true><<<gproj, blk, 0, stream>>>(
      X, Wq, nullptr, bq, Q, D, D, D, D, 0LL, 0LL, 0LL, 0LL, 1.f, 0.f);
  wmma_gemm_f32<false, false, true><<<gproj, blk, 0, stream>>>(
      X, Wk, nullptr, bk, Kb, D, D, D, D, 0LL, 0LL, 0LL, 0LL, 1.f, 0.f);
  wmma_gemm_f32<false, false, true><<<gproj, blk, 0, stream>>>(
      X, Wv, nullptr, bv, V, D, D, D, D, 0LL, 0LL, 0LL, 0LL, 1.f, 0.f);

  // 2) scores = Q_h @ K_h^T / (sqrt(hd)+1e-8) -> p0
  const float s_qk = 1.0f / (8.0f + 1e-8f);
  wmma_gemm_f32<true, false, false><<<gsc, blk, 0, stream>>>(
      Q, Kb, nullptr, nullptr, p0, HD, D, D, N, 64LL, 64LL, NN, 0LL, s_qk, 0.f);

  // 3) sim = K_h @ K_h^T / sqrt(hd) -> Pm
  wmma_gemm_f32<true, false, false><<<gsc, blk, 0, stream>>>(
      Kb, Kb, nullptr, nullptr, Pm, HD, D, D, N, 64LL, 64LL, NN, 0LL, 0.125f, 0.f);

  // 4) row softmax on both (32768 rows each)
  row_softmax<<<dim3(H * N), blk, 0, stream>>>(p0, N);
  row_softmax<<<dim3(H * N), blk, 0, stream>>>(Pm, N);

  // 5) p1 = 0.7*p0 + 0.3*(p0 @ P)  -> d_out p-region (borrowed as scratch)
  wmma_gemm_f32<false, true, false><<<gsc, blk, 0, stream>>>(
      p0, Pm, p0, nullptr, out_p, N, N, N, N, NN, NN, NN, NN, 0.3f, 0.7f);

  // 6) p2 = 0.7*p0 + 0.3*(p1 @ P)  -> in-place into p0 (Add aliases C
  //    elementwise only: each thread reads Add[idx] then writes C[idx])
  wmma_gemm_f32<false, true, false><<<gsc, blk, 0, stream>>>(
      out_p, Pm, p0, nullptr, p0, N, N, N, N, NN, NN, NN, NN, 0.3f, 0.7f);

  // 7) publish p2 into d_out p-region via the CDNA5 async LDS path
  async_copy_f32x4<<<dim3(8192), blk, 0, stream>>>(
      p0, out_p, (long long)H * NN / 4);

  // 8) attn = p2 @ V_h  (per head, N=64 columns)
  wmma_gemm_f32<false, false, false><<<gat, blk, 0, stream>>>(
      p0, V, nullptr, nullptr, At, N, N, D, D, NN, 64LL, 64LL, 0LL, 1.f, 0.f);

  // 9) out = attn @ Wo + bo
  wmma_gemm_f32<false, false, true><<<gproj, blk, 0, stream>>>(
      At, Wo, nullptr, bo, out_main, D, D, D, D, 0LL, 0LL, 0LL, 0LL, 1.f, 0.f);
}